// HGN_68478958567623
// MI455X (gfx1250) — compile-verified
//
#include <hip/hip_runtime.h>

#define NB   4096   // B
#define LSEQ 50     // L
#define DIM  64     // D
#define NT   100    // T

typedef __attribute__((ext_vector_type(2))) float v2f;
typedef __attribute__((ext_vector_type(4))) float v4f;
typedef __attribute__((ext_vector_type(8))) float v8f;

__device__ __forceinline__ float sigmoidf_(float x) {
    return 1.0f / (1.0f + __expf(-x));
}

__launch_bounds__(128)
__global__ void hgn_kernel(
    const int*   __restrict__ item_seq,   // [B, L]
    const int*   __restrict__ user_ids,   // [B]
    const int*   __restrict__ items_pred, // [B, T]
    const float* __restrict__ user_tab,   // [U, D]
    const float* __restrict__ item_tab,   // [I, D]
    const float* __restrict__ fg_item_W,  // [D, D]
    const float* __restrict__ fg_item_b,  // [D]
    const float* __restrict__ fg_user_W,  // [D, D]
    const float* __restrict__ fg_user_b,  // [D]
    const float* __restrict__ ig_item,    // [D, 1]
    const float* __restrict__ ig_user,    // [D, L]
    const float* __restrict__ W2_tab,     // [I, D]
    const float* __restrict__ b2_tab,     // [I, 1]
    float*       __restrict__ out)        // [B, T]
{
    __shared__ float sA[64 * DIM];                 // item_embs, M padded 50 -> 64 (pad rows zero)
    __shared__ alignas(16) float sWf[16 * 2 * 32 * 4]; // fg_item_W, WMMA-fragment-swizzled
    __shared__ float sG[64 * DIM];                 // gated_item
    __shared__ float sU[DIM];                      // user embedding
    __shared__ float sBias[DIM];                   // fg_item_b + fg_user_b + u @ fg_user_W
    __shared__ float sScore[LSEQ];                 // instance scores
    __shared__ float sV[DIM];                      // user + union_out + sum_l item_embs
    __shared__ int   sSeq[LSEQ];
    __shared__ float sSum;

    const int bidx = blockIdx.x;
    const int tid  = threadIdx.x;
    const int wave = tid >> 5;
    const int lane = tid & 31;

    // ---------- gather indices + user embedding ----------
    if (tid < LSEQ) sSeq[tid] = item_seq[bidx * LSEQ + tid];
    if (tid < DIM) {
        const int uid = user_ids[bidx];
        sU[tid] = user_tab[(size_t)uid * DIM + tid];
    }
    __syncthreads();

    // ---------- shared bias vector: fg_item_b + fg_user_b + u @ fg_user_W ----------
    if (tid < DIM) {
        float acc = fg_item_b[tid] + fg_user_b[tid];
        #pragma unroll 8
        for (int k = 0; k < DIM; ++k) acc += sU[k] * fg_user_W[k * DIM + tid];
        sBias[tid] = acc;
    }

    // ---------- gather item embeddings into LDS (coalesced), zero-pad rows 50..63 ----------
    for (int e = tid; e < 64 * DIM; e += 128) {
        const int m = e >> 6;
        const int d = e & 63;
        float v = 0.0f;
        if (m < LSEQ) v = item_tab[(size_t)sSeq[m] * DIM + d];
        sA[e] = v;
    }

    // ---------- stage fg_item_W in WMMA-fragment layout ----------
    // float4 slot (kk, ntp, lane) holds { B[k0][n0], B[k0+1][n0], B[k0][n1], B[k0+1][n1] }
    // with k0 = 4*kk + 2*(lane>>4), n{0,1} = (2*ntp + {0,1})*16 + (lane&15).
    // Inner loop then needs only 2x ds_load_b128 per K-step, fragments pre-aligned.
    for (int f = tid; f < 2048; f += 128) {
        const int pi    = f & 1;
        const int lane_ = (f >> 1) & 31;
        const int ntp   = (f >> 6) & 1;
        const int kk    = f >> 7;
        const int k0    = kk * 4 + ((lane_ >> 4) << 1);
        const int n     = (ntp * 2 + pi) * 16 + (lane_ & 15);
        const int dst   = (((kk * 2 + ntp) * 32 + lane_) << 2) + (pi << 1);
        sWf[dst]     = fg_item_W[k0 * DIM + n];
        sWf[dst + 1] = fg_item_W[(k0 + 1) * DIM + n];
    }
    __syncthreads();

    // ---------- WMMA GEMM: gate_pre[64x64] = sA @ fg_item_W, f32 16x16x4 tiles ----------
    const int mbase = wave * 16;
    const int mrow  = mbase + (lane & 15);
    const int khalf = (lane >> 4) * 2;

    v8f acc0 = {}, acc1 = {}, acc2 = {}, acc3 = {};
    #pragma unroll
    for (int kk = 0; kk < 16; ++kk) {
        const int k0 = kk * 4 + khalf;
        v2f af;
        af.x = sA[mrow * DIM + k0];
        af.y = sA[mrow * DIM + k0 + 1];
        const v4f bLo = *reinterpret_cast<const v4f*>(&sWf[((kk * 2 + 0) * 32 + lane) << 2]);
        const v4f bHi = *reinterpret_cast<const v4f*>(&sWf[((kk * 2 + 1) * 32 + lane) << 2]);
        v2f bf0; bf0.x = bLo.x; bf0.y = bLo.y;
        v2f bf1; bf1.x = bLo.z; bf1.y = bLo.w;
        v2f bf2; bf2.x = bHi.x; bf2.y = bHi.y;
        v2f bf3; bf3.x = bHi.z; bf3.y = bHi.w;
        acc0 = __builtin_amdgcn_wmma_f32_16x16x4_f32(false, af, false, bf0, (short)0, acc0, false, false);
        acc1 = __builtin_amdgcn_wmma_f32_16x16x4_f32(false, af, false, bf1, (short)0, acc1, false, false);
        acc2 = __builtin_amdgcn_wmma_f32_16x16x4_f32(false, af, false, bf2, (short)0, acc2, false, false);
        acc3 = __builtin_amdgcn_wmma_f32_16x16x4_f32(false, af, false, bf3, (short)0, acc3, false, false);
    }

    // ---------- apply sigmoid gate + elementwise multiply straight from C layout ----------
    // C/D layout: VGPR i -> M = mbase + (lane>=16 ? 8 : 0) + i, N = ntile*16 + (lane&15)
    {
        const int mro = mbase + (lane >> 4) * 8;
        const int n0  = lane & 15;
        #pragma unroll
        for (int i = 0; i < 8; ++i) {
            const int m = mro + i;
            float g0 = sigmoidf_(acc0[i] + sBias[n0 +  0]);
            float g1 = sigmoidf_(acc1[i] + sBias[n0 + 16]);
            float g2 = sigmoidf_(acc2[i] + sBias[n0 + 32]);
            float g3 = sigmoidf_(acc3[i] + sBias[n0 + 48]);
            sG[m * DIM + n0 +  0] = sA[m * DIM + n0 +  0] * g0;
            sG[m * DIM + n0 + 16] = sA[m * DIM + n0 + 16] * g1;
            sG[m * DIM + n0 + 32] = sA[m * DIM + n0 + 32] * g2;
            sG[m * DIM + n0 + 48] = sA[m * DIM + n0 + 48] * g3;
        }
    }
    __syncthreads();

    // ---------- instance scores ----------
    if (tid < LSEQ) {
        float s = 0.0f;
        #pragma unroll 8
        for (int d = 0; d < DIM; ++d)
            s += sG[tid * DIM + d] * ig_item[d] + sU[d] * ig_user[d * LSEQ + tid];
        sScore[tid] = sigmoidf_(s);
    }
    __syncthreads();
    if (tid == 0) {
        float ss = 0.0f;
        for (int l = 0; l < LSEQ; ++l) ss += sScore[l];
        sSum = ss;
    }
    __syncthreads();

    // ---------- combined vector v = user + union_out + sum_l item_embs ----------
    if (tid < DIM) {
        const float inv = 1.0f / sSum;
        float uo = 0.0f, si = 0.0f;
        #pragma unroll 5
        for (int l = 0; l < LSEQ; ++l) {
            uo += sG[l * DIM + tid] * sScore[l];
            si += sA[l * DIM + tid];
        }
        sV[tid] = sU[tid] + uo * inv + si;
    }
    __syncthreads();

    // ---------- scoring: wave per candidate item, wave32 butterfly reduce ----------
    for (int t = wave; t < NT; t += 4) {
        const int idx = items_pred[bidx * NT + t];
        if (t + 4 < NT) {
            const int nidx = items_pred[bidx * NT + t + 4];
            __builtin_prefetch(W2_tab + (size_t)nidx * DIM, 0, 0);  // global_prefetch_b8
        }
        const float2* w2 = reinterpret_cast<const float2*>(W2_tab + (size_t)idx * DIM);
        const float2  w  = w2[lane];
        float p = w.x * sV[2 * lane] + w.y * sV[2 * lane + 1];
        #pragma unroll
        for (int off = 16; off > 0; off >>= 1)
            p += __shfl_xor(p, off, 32);
        if (lane == 0) out[bidx * NT + t] = b2_tab[idx] + p;
    }
}

extern "C" void kernel_launch(void* const* d_in, const int* in_sizes, int n_in,
                              void* d_out, int out_size, void* d_ws, size_t ws_size,
                              hipStream_t stream) {
    const int*   item_seq   = (const int*)  d_in[0];
    const int*   user_ids   = (const int*)  d_in[1];
    const int*   items_pred = (const int*)  d_in[2];
    const float* user_tab   = (const float*)d_in[3];
    const float* item_tab   = (const float*)d_in[4];
    const float* fg_item_W  = (const float*)d_in[5];
    const float* fg_item_b  = (const float*)d_in[6];
    const float* fg_user_W  = (const float*)d_in[7];
    const float* fg_user_b  = (const float*)d_in[8];
    const float* ig_item    = (const float*)d_in[9];
    const float* ig_user    = (const float*)d_in[10];
    const float* W2_tab     = (const float*)d_in[11];
    const float* b2_tab     = (const float*)d_in[12];
    float* out = (float*)d_out;

    hgn_kernel<<<dim3(NB), dim3(128), 0, stream>>>(
        item_seq, user_ids, items_pred, user_tab, item_tab,
        fg_item_W, fg_item_b, fg_user_W, fg_user_b,
        ig_item, ig_user, W2_tab, b2_tab, out);
}